// ppo_policy_27565100106153
// MI455X (gfx1250) — compile-verified
//
#include <hip/hip_runtime.h>
#include <hip/hip_bf16.h>

typedef float v2f __attribute__((ext_vector_type(2)));
typedef float v8f __attribute__((ext_vector_type(8)));

#define G_GRAPHS 1024

// ---------------------------------------------------------------- utilities
__global__ void zero_kernel(float* __restrict__ p, long long n) {
    long long i = (long long)blockIdx.x * blockDim.x + threadIdx.x;
    if (i < n) p[i] = 0.0f;
}

__global__ void fill1_kernel(float* __restrict__ p, int n) {
    int i = blockIdx.x * blockDim.x + threadIdx.x;
    if (i < n) p[i] = 1.0f;
}

__global__ void deg_accum_kernel(const long long* __restrict__ dst,
                                 float* __restrict__ deg, int nedges) {
    int e = blockIdx.x * blockDim.x + threadIdx.x;
    if (e < nedges) atomicAdd(&deg[(int)dst[e]], 1.0f);
}

__global__ void dinv_kernel(float* __restrict__ deg, int n) {
    int i = blockIdx.x * blockDim.x + threadIdx.x;
    if (i < n) deg[i] = rsqrtf(deg[i]);   // deg >= 1 (self loop)
}

// ---------------------------------------------------- WMMA fp32 GEMM  H = X @ W
// X: [nrows, K] row-major, W: [K, NOUT] row-major, H: [nrows, NOUT]
// One wave32 computes MT 16-row tiles; NOUT covered by 16-col tiles.
// A loads are unconditional (row clamped) so the hot loop is b64 loads + wmma.
template <int K, int NOUT, int MT>
__launch_bounds__(128)
__global__ void gemm_wmma_f32(const float* __restrict__ X,
                              const float* __restrict__ W,
                              float* __restrict__ H, int nrows) {
    constexpr int KP = (K + 3) & ~3;        // K padded to x4 (zero filled)
    constexpr int NT = (NOUT + 15) / 16;    // 16-col tiles
    constexpr int KMAIN = K & ~3;           // quads fully inside K
    __shared__ float wlds[KP * 32];

    const int tid = threadIdx.x;
    for (int i = tid; i < KP * 32; i += blockDim.x) {
        int k = i >> 5, n = i & 31;
        wlds[i] = (k < K && n < NOUT) ? W[k * NOUT + n] : 0.0f;
    }
    __syncthreads();

    const int lane  = tid & 31;
    const int wave  = tid >> 5;
    const int tile0 = (blockIdx.x * (blockDim.x >> 5) + wave) * MT;
    const int m0    = tile0 * 16;
    if (m0 >= nrows) return;                 // uniform per wave: EXEC stays full

    const int mr    = lane & 15;             // A: M row / B,store: N col
    const int khalf = (lane >> 4) * 2;       // 0 or 2

    // clamped row per M-tile -> unconditional loads (bad rows discarded at store)
    int rowc[MT];
    for (int mt = 0; mt < MT; ++mt) {
        int r = m0 + mt * 16 + mr;
        rowc[mt] = (r < nrows) ? r : (nrows - 1);
    }

    v8f acc[MT][NT];
    for (int mt = 0; mt < MT; ++mt)
        for (int t = 0; t < NT; ++t) { v8f z = {}; acc[mt][t] = z; }

    for (int kk = 0; kk < KMAIN; kk += 4) {
        const int k0 = kk + khalf;
        v2f b[NT];
        for (int t = 0; t < NT; ++t) {
            b[t].x = wlds[(k0    ) * 32 + t * 16 + mr];
            b[t].y = wlds[(k0 + 1) * 32 + t * 16 + mr];
        }
        for (int mt = 0; mt < MT; ++mt) {
            v2f a = *(const v2f*)(X + (long long)rowc[mt] * K + k0); // 8B aligned
            for (int t = 0; t < NT; ++t)
                acc[mt][t] = __builtin_amdgcn_wmma_f32_16x16x4_f32(
                    false, a, false, b[t], (short)0, acc[mt][t], false, false);
        }
    }
    if (KMAIN < K) {                         // K % 4 == 2 tail (e.g. K = 54)
        const int k0 = KMAIN + khalf;
        v2f b[NT];
        for (int t = 0; t < NT; ++t) {       // LDS is zero padded to KP
            b[t].x = wlds[(k0    ) * 32 + t * 16 + mr];
            b[t].y = wlds[(k0 + 1) * 32 + t * 16 + mr];
        }
        for (int mt = 0; mt < MT; ++mt) {
            v2f a;
            a.x = (k0     < K) ? X[(long long)rowc[mt] * K + k0]     : 0.0f;
            a.y = (k0 + 1 < K) ? X[(long long)rowc[mt] * K + k0 + 1] : 0.0f;
            for (int t = 0; t < NT; ++t)
                acc[mt][t] = __builtin_amdgcn_wmma_f32_16x16x4_f32(
                    false, a, false, b[t], (short)0, acc[mt][t], false, false);
        }
    }

    const int mofs = (lane >> 4) * 8;        // C/D: vgpr j -> M = j + 8*(lane>=16)
    for (int mt = 0; mt < MT; ++mt)
        for (int t = 0; t < NT; ++t)
            for (int j = 0; j < 8; ++j) {
                int m = m0 + mt * 16 + mofs + j;
                int n = t * 16 + mr;
                if (m < nrows && n < NOUT)
                    H[(long long)m * NOUT + n] = acc[mt][t][j];
            }
}

// ------------------------------------------------------------- edge scatter
// F/4 lanes per edge, float4 gathers; atomics are per-float (no vector atomic).
template <int F>
__global__ void gcn_scatter(const long long* __restrict__ src,
                            const long long* __restrict__ dst,
                            const float* __restrict__ dinv,
                            const float* __restrict__ h,
                            float* __restrict__ acc, int nedges) {
    constexpr int LPE = F / 4;
    long long tid = (long long)blockIdx.x * blockDim.x + threadIdx.x;
    int e   = (int)(tid / LPE);
    int sub = (int)(tid % LPE);
    if (e >= nedges) return;
    int s = (int)src[e];
    int d = (int)dst[e];
    float norm = dinv[s] * dinv[d];
    float4 hv = *(const float4*)(h + (long long)s * F + sub * 4);
    float* ap = acc + (long long)d * F + sub * 4;
    atomicAdd(ap + 0, hv.x * norm);
    atomicAdd(ap + 1, hv.y * norm);
    atomicAdd(ap + 2, hv.z * norm);
    atomicAdd(ap + 3, hv.w * norm);
}

// self-loop + bias + relu, in place on acc
template <int F>
__global__ void gcn_finish(float* __restrict__ acc,
                           const float* __restrict__ h,
                           const float* __restrict__ dinv,
                           const float* __restrict__ bias, int nrows) {
    long long tid = (long long)blockIdx.x * blockDim.x + threadIdx.x;
    if (tid >= (long long)nrows * F) return;
    int i = (int)(tid / F);
    int f = (int)(tid % F);
    float di = dinv[i];
    float v = acc[tid] + h[tid] * di * di + bias[f];
    acc[tid] = fmaxf(v, 0.0f);
}

// ----------------------------------------------------------------- pooling
__global__ void cnt_kernel(const long long* __restrict__ batch,
                           float* __restrict__ cnt, int nrows) {
    int i = blockIdx.x * blockDim.x + threadIdx.x;
    if (i < nrows) atomicAdd(&cnt[(int)batch[i]], 1.0f);
}

// sum[g, f] += x[i, f] * (scale ? scale[i] : 1)  (F = 24, 6 lanes/node, float4)
__global__ void pool24_kernel(const long long* __restrict__ batch,
                              const float* __restrict__ x,
                              const float* __restrict__ scale,
                              float* __restrict__ sum, int nrows) {
    long long tid = (long long)blockIdx.x * blockDim.x + threadIdx.x;
    int i   = (int)(tid / 6);
    int sub = (int)(tid % 6);
    if (i >= nrows) return;
    int g = (int)batch[i];
    float4 v = *(const float4*)(x + (long long)i * 24 + sub * 4);
    if (scale) {
        float sc = scale[i];
        v.x *= sc; v.y *= sc; v.z *= sc; v.w *= sc;
    }
    float* sp = sum + g * 24 + sub * 4;
    atomicAdd(sp + 0, v.x);
    atomicAdd(sp + 1, v.y);
    atomicAdd(sp + 2, v.z);
    atomicAdd(sp + 3, v.w);
}

__global__ void mean24_kernel(float* __restrict__ sum,
                              const float* __restrict__ cnt, int total) {
    int tid = blockIdx.x * blockDim.x + threadIdx.x;
    if (tid >= total) return;
    int g = tid / 24;
    sum[tid] /= fmaxf(cnt[g], 1.0f);
}

// ----------------------------------------- dense: one thread per row (N-sized)
template <int FI, int FO, int RELU>
__global__ void dense_rows(const float* __restrict__ in,
                           const float* __restrict__ W,
                           const float* __restrict__ bias,
                           float* __restrict__ out, int rows) {
    int r = blockIdx.x * blockDim.x + threadIdx.x;
    if (r >= rows) return;
    float xi[FI];
    for (int k = 0; k < FI; ++k) xi[k] = in[(long long)r * FI + k];
    for (int c = 0; c < FO; ++c) {
        float s = bias[c];
        for (int k = 0; k < FI; ++k) s += xi[k] * W[k * FO + c];
        out[(long long)r * FO + c] = RELU ? fmaxf(s, 0.0f) : s;
    }
}

// ------------------------------------------- dense: one thread per element (G)
__global__ void dense_kernel(const float* __restrict__ in,
                             const float* __restrict__ W,
                             const float* __restrict__ bias,
                             float* __restrict__ out,
                             int rows, int FI, int FO, int dorelu) {
    int tid = blockIdx.x * blockDim.x + threadIdx.x;
    if (tid >= rows * FO) return;
    int r = tid / FO, c = tid % FO;
    float s = bias[c];
    const float* xi = in + (long long)r * FI;
    for (int k = 0; k < FI; ++k) s += xi[k] * W[k * FO + c];
    out[tid] = dorelu ? fmaxf(s, 0.0f) : s;
}

// ---------------------------------------------------------------- launcher
extern "C" void kernel_launch(void* const* d_in, const int* in_sizes, int n_in,
                              void* d_out, int out_size, void* d_ws, size_t ws_size,
                              hipStream_t stream) {
    const float*     x    = (const float*)d_in[0];
    const long long* ei   = (const long long*)d_in[1];
    const long long* bat  = (const long long*)d_in[2];
    const float *w1 = (const float*)d_in[4],  *b1 = (const float*)d_in[5];
    const float *w2 = (const float*)d_in[6],  *b2 = (const float*)d_in[7];
    const float *w3 = (const float*)d_in[8],  *b3 = (const float*)d_in[9];
    const float *gw1 = (const float*)d_in[10], *gb1 = (const float*)d_in[11];
    const float *gw2 = (const float*)d_in[12], *gb2 = (const float*)d_in[13];
    const float *tw  = (const float*)d_in[14], *tb  = (const float*)d_in[15];
    const float *nw1 = (const float*)d_in[16], *nb1 = (const float*)d_in[17];
    const float *nw2 = (const float*)d_in[18], *nb2 = (const float*)d_in[19];
    const float *bw1 = (const float*)d_in[20], *bb1 = (const float*)d_in[21];
    const float *bw2 = (const float*)d_in[22], *bb2 = (const float*)d_in[23];

    const int N = in_sizes[2];            // 150000 (batch length)
    const int E = in_sizes[1] / 2;        // 2400000
    const int G = G_GRAPHS;               // 1024
    const long long* src = ei;
    const long long* dst = ei + E;

    // workspace layout (floats)
    float* ws   = (float*)d_ws;
    float* dinv = ws;                      ws += N;
    float* A    = ws;                      ws += (long long)N * 32;  // gemm out
    float* B    = ws;                      ws += (long long)N * 32;  // acc / x
    float* C    = ws;                      ws += (long long)N * 32;  // acc / x
    float* cnt  = ws;                      ws += G;
    float* gsum = ws;                      ws += G * 24;
    float* g1   = ws;                      ws += G * 24;
    float* g2   = ws;                      ws += G * 32;
    float* bsum = ws;                      ws += G * 24;
    float* bg1  = ws;                      ws += G * 16;

    float* out_t = (float*)d_out;                  // [G, 2]
    float* out_n = out_t + (long long)G * 2;       // [N, 1]
    float* out_b = out_n + N;                      // [G, 3]

    auto cdiv = [](long long a, long long b) { return (int)((a + b - 1) / b); };
    const int TB = 256;
    constexpr int MT = 4;                          // M tiles per wave
    const int gemm_blocks = cdiv(cdiv(N, 16 * MT), 4);   // 4 waves / block

    // ---- degree norm
    fill1_kernel<<<cdiv(N, TB), TB, 0, stream>>>(dinv, N);
    deg_accum_kernel<<<cdiv(E, TB), TB, 0, stream>>>(dst, dinv, E);
    dinv_kernel<<<cdiv(N, TB), TB, 0, stream>>>(dinv, N);

    // ---- conv1: 54 -> 32
    gemm_wmma_f32<54, 32, MT><<<gemm_blocks, 128, 0, stream>>>(x, w1, A, N);
    zero_kernel<<<cdiv((long long)N * 32, TB), TB, 0, stream>>>(B, (long long)N * 32);
    gcn_scatter<32><<<cdiv((long long)E * 8, TB), TB, 0, stream>>>(src, dst, dinv, A, B, E);
    gcn_finish<32><<<cdiv((long long)N * 32, TB), TB, 0, stream>>>(B, A, dinv, b1, N);

    // ---- conv2: 32 -> 32
    gemm_wmma_f32<32, 32, MT><<<gemm_blocks, 128, 0, stream>>>(B, w2, A, N);
    zero_kernel<<<cdiv((long long)N * 32, TB), TB, 0, stream>>>(C, (long long)N * 32);
    gcn_scatter<32><<<cdiv((long long)E * 8, TB), TB, 0, stream>>>(src, dst, dinv, A, C, E);
    gcn_finish<32><<<cdiv((long long)N * 32, TB), TB, 0, stream>>>(C, A, dinv, b2, N);

    // ---- conv3: 32 -> 24   (x3 lands in B)
    gemm_wmma_f32<32, 24, MT><<<gemm_blocks, 128, 0, stream>>>(C, w3, A, N);
    zero_kernel<<<cdiv((long long)N * 24, TB), TB, 0, stream>>>(B, (long long)N * 24);
    gcn_scatter<24><<<cdiv((long long)E * 6, TB), TB, 0, stream>>>(src, dst, dinv, A, B, E);
    gcn_finish<24><<<cdiv((long long)N * 24, TB), TB, 0, stream>>>(B, A, dinv, b3, N);
    float* x3 = B;

    // ---- pooled graph head: t
    zero_kernel<<<cdiv(G, TB), TB, 0, stream>>>(cnt, G);
    zero_kernel<<<cdiv((long long)G * 24, TB), TB, 0, stream>>>(gsum, (long long)G * 24);
    cnt_kernel<<<cdiv(N, TB), TB, 0, stream>>>(bat, cnt, N);
    pool24_kernel<<<cdiv((long long)N * 6, TB), TB, 0, stream>>>(bat, x3, nullptr, gsum, N);
    mean24_kernel<<<cdiv(G * 24, TB), TB, 0, stream>>>(gsum, cnt, G * 24);
    dense_kernel<<<cdiv(G * 24, TB), TB, 0, stream>>>(gsum, gw1, gb1, g1, G, 24, 24, 0);
    dense_kernel<<<cdiv(G * 32, TB), TB, 0, stream>>>(g1, gw2, gb2, g2, G, 24, 32, 0);
    dense_kernel<<<cdiv(G * 2, TB), TB, 0, stream>>>(g2, tw, tb, out_t, G, 32, 2, 1);

    // ---- node head: n   (n1 in C)
    dense_rows<24, 16, 1><<<cdiv(N, TB), TB, 0, stream>>>(x3, nw1, nb1, C, N);
    dense_rows<16, 1, 1><<<cdiv(N, TB), TB, 0, stream>>>(C, nw2, nb2, out_n, N);

    // ---- b head: weighted add-pool of n * x3
    zero_kernel<<<cdiv((long long)G * 24, TB), TB, 0, stream>>>(bsum, (long long)G * 24);
    pool24_kernel<<<cdiv((long long)N * 6, TB), TB, 0, stream>>>(bat, x3, out_n, bsum, N);
    dense_kernel<<<cdiv(G * 16, TB), TB, 0, stream>>>(bsum, bw1, bb1, bg1, G, 24, 16, 1);
    dense_kernel<<<cdiv(G * 3, TB), TB, 0, stream>>>(bg1, bw2, bb2, out_b, G, 16, 3, 1);
}